// ROI_Relation_79448305041449
// MI455X (gfx1250) — compile-verified
//
#include <hip/hip_runtime.h>
#include <hip/hip_bf16.h>
#include <stddef.h>
#include <stdint.h>

// ---------------------------------------------------------------------------
// Problem constants (match the reference)
// ---------------------------------------------------------------------------
#define BB   4
#define TT   2048
#define DD   512
#define RR   1024
#define NBIN 13            // 2 start + 9 inner + 2 end
#define MTOT (BB * RR)     // 4096 rows
#define PSTR (DD * NBIN)   // 6656 floats per pooled row

typedef float v2f __attribute__((ext_vector_type(2)));
typedef float v8f __attribute__((ext_vector_type(8)));

__device__ __forceinline__ float selu_f(float x) {
    const float scale = 1.0507009873554805f;
    const float alpha = 1.6732632423543772f;
    return x > 0.0f ? scale * x : scale * alpha * (expf(x) - 1.0f);
}

// CDNA5 async global->LDS copy (4B per lane), tracked by ASYNCcnt.
// Generic pointers to __shared__ carry the LDS offset in their low 32 bits.
__device__ __forceinline__ void async_copy_b32(void* lds_dst, const void* gsrc) {
    unsigned           ldsoff = (unsigned)(uintptr_t)lds_dst;
    unsigned long long gaddr  = (unsigned long long)(uintptr_t)gsrc;
    asm volatile("global_load_async_to_lds_b32 %0, %1, off"
                 :: "v"(ldsoff), "v"(gaddr) : "memory");
}
__device__ __forceinline__ void wait_async0() {
    asm volatile("s_wait_asynccnt 0x0" ::: "memory");
}

// ---------------------------------------------------------------------------
// Kernel 1: BRoI1D pooling.  One block per ROI row m; threads sweep D.
// pooled layout: [m][d][bin] (bin fastest) -> FC slices are strided sub-views.
// ---------------------------------------------------------------------------
__global__ void __launch_bounds__(256)
pool_kernel(const float* __restrict__ feats,   // (B, T, D)
            const float* __restrict__ rois,    // (B, R, 2)
            float* __restrict__ pooled)        // (MTOT, D, NBIN)
{
    const int m = blockIdx.x;          // 0..4095
    const int b = m >> 10;             // m / R
    const float s = rois[(size_t)m * 2 + 0];
    const float e = rois[(size_t)m * 2 + 1];
    const float ext = (1.0f / 5.0f) * (e - s);

    int   lo[NBIN], hi[NBIN];
    float w[NBIN];
#pragma unroll
    for (int k = 0; k < NBIN; ++k) {
        float pos;
        if (k < 2)        pos = (s - ext) + ((float)k + 0.5f) * ext;
        else if (k < 11)  pos = s + (((float)(k - 2) + 0.5f) / 9.0f) * (e - s);
        else              pos = (e - ext) + ((float)(k - 11) + 0.5f) * ext;
        pos = fminf(fmaxf(pos, 0.0f), (float)(TT - 1));
        int l = (int)floorf(pos);
        lo[k] = l;
        hi[k] = min(l + 1, TT - 1);
        w[k]  = pos - (float)l;
    }

    const float* fb = feats + (size_t)b * TT * DD;
    float* prow = pooled + (size_t)m * PSTR;

    for (int d = threadIdx.x; d < DD; d += blockDim.x) {
#pragma unroll
        for (int k = 0; k < NBIN; ++k) {
            float vlo = fb[(size_t)lo[k] * DD + d];
            float vhi = fb[(size_t)hi[k] * DD + d];
            prow[d * NBIN + k] = (1.0f - w[k]) * vlo + w[k] * vhi;
        }
    }
}

// ---------------------------------------------------------------------------
// Kernel 2: z = [right - left, inner]   (4096 x 1024)
// ---------------------------------------------------------------------------
__global__ void __launch_bounds__(256)
diff_kernel(const float* __restrict__ leftb,
            const float* __restrict__ rightb,
            const float* __restrict__ innerb,
            float* __restrict__ z)
{
    int idx = blockIdx.x * 256 + threadIdx.x;   // 0 .. 4M-1
    int m = idx >> 10;
    int j = idx & 1023;
    size_t base = (size_t)m * DD;
    z[idx] = (j < DD) ? (rightb[base + j] - leftb[base + j])
                      : innerb[base + j - DD];
}

// ---------------------------------------------------------------------------
// Kernel 3: SELU( A @ W^T + bias ) GEMM with V_WMMA_F32_16X16X4_F32.
//   M=4096, N=512, BM=128 x BN=64 x BK=16, 256 threads (8 waves).
//   Double-buffered LDS, async global->LDS staging overlapped with WMMA.
//
// A mapping: col = (nb>0) ? (i/nb)*13 + bin0 + i%nb : i ; element A[m*lda+col]
//
// Fragment layouts (ISA 7.12.2, 32-bit):
//   A 16x4:  lanes 0-15 -> {K=k0,k0+1}; lanes 16-31 -> {k0+2,k0+3}
//   B 4x16:  vreg j holds row K=k0+j (lanes 0-15) / K=k0+2+j (lanes 16-31)
//   C 16x16: vreg v -> M = v / v+8 per lane half, N = lane%16
// LDS pads: A row = 18 floats (72B, even-k pairs 8B aligned, conflict-light);
//           B row = 20 floats (80B, 16B aligned, conflict-free b64 frags).
// ---------------------------------------------------------------------------
#define BM 128
#define BN 64
#define BK 16
#define ASTR 18
#define BSTR 20

__global__ void __launch_bounds__(256)
gemm_selu_kernel(const float* __restrict__ A, int lda, int nb, int bin0,
                 const float* __restrict__ W,     // (512, K) row-major
                 const float* __restrict__ bias,  // (512,)
                 float* __restrict__ out,         // (4096, 512) row-major
                 int K)
{
    __shared__ __align__(16) float As[2][BM * ASTR];
    __shared__ __align__(16) float Bs[2][BN * BSTR];

    const int tid  = threadIdx.x;
    const int wave = tid >> 5;
    const int lane = tid & 31;
    const int half = lane >> 4;
    const int l    = lane & 15;
    const int m0   = blockIdx.x * BM;
    const int n0   = blockIdx.y * BN;

    v8f acc[4];
#pragma unroll
    for (int t = 0; t < 4; ++t) acc[t] = (v8f){};

    // ---- async staging of one (BMxBK, BNxBK) tile pair ---------------------
    auto stage = [&](int kt, int bufi) {
        float* Asb = &As[bufi][0];
        float* Bsb = &Bs[bufi][0];
#pragma unroll
        for (int u = 0; u < 8; ++u) {              // A: 2048 elems, 8/thread
            int idx = tid + u * 256;
            int row = idx >> 4;
            int kk  = idx & 15;
            int i   = kt + kk;
            int col;
            if (nb > 0) { int d = i / nb; col = d * NBIN + bin0 + (i - d * nb); }
            else        col = i;
            async_copy_b32(&Asb[row * ASTR + kk],
                           A + (size_t)(m0 + row) * lda + col);
        }
#pragma unroll
        for (int u = 0; u < 4; ++u) {              // B: 1024 elems, 4/thread
            int idx = tid + u * 256;
            int n   = idx >> 4;
            int kk  = idx & 15;
            async_copy_b32(&Bsb[n * BSTR + kk],
                           W + (size_t)(n0 + n) * K + kt + kk);
        }
    };

    auto compute = [&](int bufi) {
        const float* Asb = &As[bufi][0];
        const float* Bsb = &Bs[bufi][0];
#pragma unroll
        for (int c4 = 0; c4 < 4; ++c4) {
            const int k0 = c4 * 4 + half * 2;
            v2f a = *(const v2f*)(Asb + (wave * 16 + l) * ASTR + k0);
#pragma unroll
            for (int t = 0; t < 4; ++t) {
                v2f bf = *(const v2f*)(Bsb + (t * 16 + l) * BSTR + k0);
                acc[t] = __builtin_amdgcn_wmma_f32_16x16x4_f32(
                    false, a, false, bf, (short)0, acc[t], false, false);
            }
        }
    };

    const int nt = K / BK;
    stage(0, 0);
    wait_async0();
    __syncthreads();
    for (int t = 0; t < nt; ++t) {
        const int cur = t & 1;
        if (t + 1 < nt) stage((t + 1) * BK, cur ^ 1);   // prefetch next tile
        compute(cur);                                    // WMMA on current
        wait_async0();                                   // our DMA done
        __syncthreads();                                 // everyone done
    }

    // ---- epilogue: bias + SELU -------------------------------------------
#pragma unroll
    for (int t = 0; t < 4; ++t) {
        const float* cp = (const float*)&acc[t];
        int gn = n0 + t * 16 + l;
        float bv = bias[gn];
#pragma unroll
        for (int v = 0; v < 8; ++v) {
            int gm = m0 + wave * 16 + half * 8 + v;
            out[(size_t)gm * DD + gn] = selu_f(cp[v] + bv);
        }
    }
}

// ---------------------------------------------------------------------------
// Host-side orchestration (graph-capture safe: only async launches on stream)
// ---------------------------------------------------------------------------
extern "C" void kernel_launch(void* const* d_in, const int* in_sizes, int n_in,
                              void* d_out, int out_size, void* d_ws, size_t ws_size,
                              hipStream_t stream) {
    (void)in_sizes; (void)n_in; (void)out_size; (void)ws_size;

    const float* features = (const float*)d_in[0];   // (B, T, D)
    const float* rois     = (const float*)d_in[3];   // (B, R, 2)
    const float* W_left   = (const float*)d_in[6];   // (512, 3584)
    const float* b_left   = (const float*)d_in[7];
    const float* W_inner  = (const float*)d_in[8];   // (512, 4608)
    const float* b_inner  = (const float*)d_in[9];
    const float* W_roi    = (const float*)d_in[10];  // (512, 1024)
    const float* b_roi    = (const float*)d_in[11];

    // Workspace layout (144 MB total):
    //   pooled : 4096*6656 f32          (104 MB)
    //   left/right/inner : 3 x 4096*512 (24 MB)
    //   z : 4096*1024                   (16 MB)
    float* pooled = (float*)d_ws;
    float* leftb  = pooled + (size_t)MTOT * PSTR;
    float* rightb = leftb  + (size_t)MTOT * DD;
    float* innerb = rightb + (size_t)MTOT * DD;
    float* zbuf   = innerb + (size_t)MTOT * DD;

    pool_kernel<<<MTOT, 256, 0, stream>>>(features, rois, pooled);

    dim3 grid(MTOT / BM, DD / BN);   // 32 x 8
    dim3 block(256);

    // left  = selu(pooled[:, :, 0:7]  @ W_left^T  + b_left)   K=3584
    gemm_selu_kernel<<<grid, block, 0, stream>>>(
        pooled, PSTR, 7, 0, W_left, b_left, leftb, 7 * DD);
    // right = selu(pooled[:, :, 6:13] @ W_left^T  + b_left)   K=3584
    gemm_selu_kernel<<<grid, block, 0, stream>>>(
        pooled, PSTR, 7, 6, W_left, b_left, rightb, 7 * DD);
    // inner = selu(pooled[:, :, 2:11] @ W_inner^T + b_inner)  K=4608
    gemm_selu_kernel<<<grid, block, 0, stream>>>(
        pooled, PSTR, 9, 2, W_inner, b_inner, innerb, 9 * DD);

    diff_kernel<<<(MTOT * 2 * DD) / 256, 256, 0, stream>>>(
        leftb, rightb, innerb, zbuf);

    // out = selu(z @ W_roi^T + b_roi)   K=1024
    gemm_selu_kernel<<<grid, block, 0, stream>>>(
        zbuf, 2 * DD, 0, 0, W_roi, b_roi, (float*)d_out, 2 * DD);
}